// GraphConvolution_7181185319265
// MI455X (gfx1250) — compile-verified
//
#include <hip/hip_runtime.h>

typedef float v2f __attribute__((ext_vector_type(2)));
typedef float v8f __attribute__((ext_vector_type(8)));

#define D 128  // D_IN == D_OUT == 128

// ---------------------------------------------------------------------------
// Kernel 1: h = x @ W^T + b using V_WMMA_F32_16X16X4_F32 (exact fp32 path).
// Block = 256 threads = 8 wave32; each wave computes a 16-row x 128-col strip.
// ---------------------------------------------------------------------------
__global__ void __launch_bounds__(256) gcn_linear_wmma(
    const float* __restrict__ x, const float* __restrict__ W,
    const float* __restrict__ b, float* __restrict__ h, int N)
{
  __shared__ float Wl[D * D];  // 64 KB: W row-major [out][in]

  // Cooperative 128-bit staging of W into LDS.
  {
    const float4* Ws = (const float4*)W;
    float4* Wd = (float4*)Wl;
    for (int i = threadIdx.x; i < (D * D) / 4; i += 256)
      Wd[i] = Ws[i];
  }
  __syncthreads();

  const int lane = threadIdx.x & 31;
  const int wave = threadIdx.x >> 5;
  const int half = lane >> 4;   // 0: lanes 0-15, 1: lanes 16-31
  const int l16  = lane & 15;
  const int m0   = blockIdx.x * 128 + wave * 16;

  // A-fragment source row (clamp OOB rows to a valid address; stores guarded).
  int row  = m0 + l16;
  int rowc = row < N ? row : (N - 1);
  const float* xr = x + (size_t)rowc * D + 2 * half;

  // 8 N-tiles of 16 cols each; init accumulators with bias (broadcast over M).
  v8f acc[8];
#pragma unroll
  for (int t = 0; t < 8; ++t) {
    float bv = b[t * 16 + l16];
    v8f ai = {bv, bv, bv, bv, bv, bv, bv, bv};
    acc[t] = ai;
  }

  // K loop: 32 steps of K=4. A frag reused across all 8 N-tiles.
#pragma unroll 4
  for (int k0 = 0; k0 < D; k0 += 4) {
    // A(16x4): lane l holds x[m0 + (l&15)][k0 + 2*half + {0,1}] -> float2
    v2f a = *(const v2f*)(xr + k0);
#pragma unroll
    for (int t = 0; t < 8; ++t) {
      // B(4x16): B[k][n] = W[n][k]; VGPR v <-> K = k0 + 2*half + v -> float2
      v2f bw = *(const v2f*)(Wl + (size_t)(t * 16 + l16) * D + k0 + 2 * half);
      acc[t] = __builtin_amdgcn_wmma_f32_16x16x4_f32(
          /*neg_a=*/false, a, /*neg_b=*/false, bw,
          /*c_mod=*/(short)0, acc[t], /*reuse_a=*/false, /*reuse_b=*/false);
    }
  }

  // C/D layout: VGPR r holds rows M=r (lanes 0-15) and M=r+8 (lanes 16-31).
#pragma unroll
  for (int t = 0; t < 8; ++t) {
    int n = t * 16 + l16;
#pragma unroll
    for (int r = 0; r < 8; ++r) {
      int m = m0 + r + 8 * half;
      if (m < N) h[(size_t)m * D + n] = acc[t][r];
    }
  }
}

// ---------------------------------------------------------------------------
// Kernel 2a: zero the (poisoned) output.
// ---------------------------------------------------------------------------
__global__ void gcn_zero(float4* __restrict__ out, int n4)
{
  int i = blockIdx.x * blockDim.x + threadIdx.x;
  if (i < n4) out[i] = make_float4(0.f, 0.f, 0.f, 0.f);
}

// ---------------------------------------------------------------------------
// Kernel 2b: scatter-add SpMM. One wave32 per edge: float4 gather of h[col]
// (L2-resident), scale by edge weight, fp32 global atomics into out[row].
// ---------------------------------------------------------------------------
__global__ void __launch_bounds__(256) gcn_spmm_scatter(
    const float* __restrict__ h, const int* __restrict__ rows,
    const int* __restrict__ cols, const float* __restrict__ vals,
    float* __restrict__ out, int E)
{
  int e    = (blockIdx.x * 256 + threadIdx.x) >> 5;
  int lane = threadIdx.x & 31;
  if (e >= E) return;

  int   rdst = rows[e];
  int   csrc = cols[e];
  float v    = vals[e];

  float4 hv = ((const float4*)(h + (size_t)csrc * D))[lane];  // 16 B/lane
  float* o  = out + (size_t)rdst * D + lane * 4;

  unsafeAtomicAdd(o + 0, hv.x * v);
  unsafeAtomicAdd(o + 1, hv.y * v);
  unsafeAtomicAdd(o + 2, hv.z * v);
  unsafeAtomicAdd(o + 3, hv.w * v);
}

// ---------------------------------------------------------------------------
extern "C" void kernel_launch(void* const* d_in, const int* in_sizes, int n_in,
                              void* d_out, int out_size, void* d_ws, size_t ws_size,
                              hipStream_t stream)
{
  const float* x    = (const float*)d_in[0];
  const int*   rows = (const int*)d_in[1];
  const int*   cols = (const int*)d_in[2];
  const float* vals = (const float*)d_in[3];
  const float* W    = (const float*)d_in[4];
  const float* b    = (const float*)d_in[5];
  float*       out  = (float*)d_out;

  const int N = in_sizes[0] / D;  // 50000
  const int E = in_sizes[1];      // 800000

  float* h = (float*)d_ws;        // N*D*4 = 25.6 MB scratch

  // Stage 1: h = x @ W^T + b
  gcn_linear_wmma<<<(N + 127) / 128, 256, 0, stream>>>(x, W, b, h, N);

  // Stage 2: out = segment_sum(vals * h[cols], rows)
  int n4 = out_size / 4;
  gcn_zero<<<(n4 + 255) / 256, 256, 0, stream>>>((float4*)out, n4);

  long long spmm_threads = (long long)E * 32;
  gcn_spmm_scatter<<<(unsigned)((spmm_threads + 255) / 256), 256, 0, stream>>>(
      h, rows, cols, vals, out, E);
}